// OnlineHadamardTransform_22746146800067
// MI455X (gfx1250) — compile-verified
//
#include <hip/hip_runtime.h>

typedef __attribute__((ext_vector_type(2))) float v2f;
typedef __attribute__((ext_vector_type(4))) float v4f;
typedef __attribute__((ext_vector_type(8))) float v8f;

#define HAD          128
#define WAVES_PER_WG 8
#define WG_THREADS   (WAVES_PER_WG * 32)
#define XSTRIDE      132                       // padded row stride (floats) -> conflict-free b64 A loads
#define H_FLOATS     (HAD * HAD)               // 16384 floats, packed B-operand layout
#define XTILE_FLOATS (16 * XSTRIDE)            // 2112 floats per wave
#define LDS_FLOATS   (H_FLOATS + WAVES_PER_WG * XTILE_FLOATS)

// y[m, :] = x[m, :] @ H  for 524288 rows of length 128, tensor-core fp32 path.
__global__ __launch_bounds__(WG_THREADS)
void had128_wmma_f32(const float* __restrict__ x,
                     const float* __restrict__ r2,
                     float* __restrict__ y,
                     int nBlocks /* total 16-row blocks */) {
    extern __shared__ float lds[];
    float* ldsH = lds;                                  // H in WMMA B layout
    const int tid  = threadIdx.x;
    const int lane = tid & 31;
    const int wave = tid >> 5;
    float* ldsX = lds + H_FLOATS + wave * XTILE_FLOATS; // per-wave x tile

    // ---- Stage H (r2) into LDS, pre-swizzled into the B-operand lane layout ----
    // flat f: j=f&1 (VGPR0/1), l=(f>>1)&31 (lane), nt=(f>>6)&7 (N tile), kt=f>>9 (K step)
    // B VGPR0 holds K = 4*kt + 2*(l>=16), VGPR1 holds K+1; N = nt*16 + (l&15).
    for (int f = tid; f < H_FLOATS; f += WG_THREADS) {
        const int j  = f & 1;
        const int l  = (f >> 1) & 31;
        const int nt = (f >> 6) & 7;
        const int kt = f >> 9;
        const int k  = kt * 4 + ((l >> 4) << 1) + j;
        const int n  = nt * 16 + (l & 15);
        ldsH[f] = r2[k * HAD + n];
    }
    __syncthreads();

    const int   arow = lane & 15;          // A-matrix row within tile
    const int   koff = (lane >> 4) << 1;   // K offset for this half-wave
    const int   waveGlobal = blockIdx.x * WAVES_PER_WG + wave;
    const int   waveCount  = gridDim.x * WAVES_PER_WG;

    for (int blk = waveGlobal; blk < nBlocks; blk += waveCount) {
        const long  m0   = (long)blk * 16;
        const float* xb  = x + m0 * HAD;
        float*       yb  = y + m0 * HAD;

        // ---- coalesced 512B row loads into padded LDS tile ----
        #pragma unroll
        for (int r = 0; r < 16; ++r) {
            v4f v = *(const v4f*)(xb + r * HAD + lane * 4);
            *(v4f*)(ldsX + r * XSTRIDE + lane * 4) = v;
        }

        v8f acc[8] = {};

        // ---- 32 K-steps x 8 N-tiles of V_WMMA_F32_16X16X4_F32 ----
        #pragma unroll 4
        for (int ks = 0; ks < 32; ++ks) {
            v2f a = *(const v2f*)(ldsX + arow * XSTRIDE + ks * 4 + koff);
            #pragma unroll
            for (int n = 0; n < 8; ++n) {
                v2f b = *(const v2f*)(ldsH + ((((ks * 8 + n) * 32) + lane) << 1));
                acc[n] = __builtin_amdgcn_wmma_f32_16x16x4_f32(
                    /*neg_a=*/false, a, /*neg_b=*/false, b,
                    /*c_mod=*/(short)0, acc[n],
                    /*reuse_a=*/false, /*reuse_b=*/false);
            }
        }

        // ---- repack D through LDS so the global store is coalesced 512B rows ----
        // C/D layout: VGPR v, lanes 0-15 -> (M=v, N=lane), lanes 16-31 -> (M=v+8, N=lane-16)
        #pragma unroll
        for (int n = 0; n < 8; ++n) {
            #pragma unroll
            for (int v = 0; v < 8; ++v) {
                const int row = v + ((lane >> 4) << 3);
                ldsX[row * XSTRIDE + n * 16 + (lane & 15)] = acc[n][v];
            }
        }
        #pragma unroll
        for (int r = 0; r < 16; ++r) {
            v4f v = *(const v4f*)(ldsX + r * XSTRIDE + lane * 4);
            *(v4f*)(yb + r * HAD + lane * 4) = v;
        }
    }
}

extern "C" void kernel_launch(void* const* d_in, const int* in_sizes, int n_in,
                              void* d_out, int out_size, void* d_ws, size_t ws_size,
                              hipStream_t stream) {
    const float* x  = (const float*)d_in[0];
    const float* r2 = (const float*)d_in[1];
    float*       y  = (float*)d_out;

    const int rows    = in_sizes[0] / HAD;   // 524288
    const int nBlocks = rows / 16;           // 32768 (rows divisible by 16)
    const int grid    = 512;                 // grid-stride: ~8 blocks per wave, amortizes H staging
    const size_t shmem = LDS_FLOATS * sizeof(float); // ~130 KB (<= 320 KB/WGP)

    hipFuncSetAttribute(reinterpret_cast<const void*>(had128_wmma_f32),
                        hipFuncAttributeMaxDynamicSharedMemorySize, (int)shmem);

    had128_wmma_f32<<<grid, WG_THREADS, shmem, stream>>>(x, r2, y, nBlocks);
}